// GlobalAggregator_68453188764261
// MI455X (gfx1250) — compile-verified
//
#include <hip/hip_runtime.h>

// Problem constants (from reference): B=512, N=64, K=12, DIM=128
#define BB   512
#define NNN  64
#define KNB  12
#define DIM  128
#define BN   (BB * NNN)          // 32768 independent (b,n) rows

typedef float v2f __attribute__((ext_vector_type(2)));
typedef float v8f __attribute__((ext_vector_type(8)));

// ---------------- Kernel 1 LDS layout (floats) ----------------
#define W1_STRIDE   132                       // padded row stride for w1 tile (528 B, 16B-aligned)
#define W1_FLOATS   (132 * W1_STRIDE)         // rows 0..128 real, 129..131 zero
#define NB_STRIDE   132                       // neighbor tile row stride
#define NB_FLOATS   (16 * NB_STRIDE)          // rows 12..15 zero padding
#define WAVE_FLOATS (NB_FLOATS + 132 + 128)   // nb tile + extra(132) + w2(128) = 2372
#define K1_WAVES    8
#define K1_LDS_FLOATS (W1_FLOATS + K1_WAVES * WAVE_FLOATS)   // 36400 floats (145.6 KB)

// ---------------- Kernel 2 LDS layout ----------------
#define K2_A_STRIDE 260                       // 16 rows x 256 cols (self||agg); 1040 B rows, 16B-aligned
#define K2_LDS_FLOATS (16 * K2_A_STRIDE)

__device__ __forceinline__ v8f wmma_f32_4(v2f a, v2f b, v8f c) {
    return __builtin_amdgcn_wmma_f32_16x16x4_f32(
        /*neg_a=*/false, a, /*neg_b=*/false, b,
        /*c_mod=*/(short)0, c, /*reuse_a=*/false, /*reuse_b=*/false);
}

// gfx1250 async global->LDS copy (bypasses VGPRs, tracked by ASYNCcnt).
__device__ __forceinline__ void async_ld_b128(unsigned lds_byte_off, const float* gaddr) {
    asm volatile("global_load_async_to_lds_b128 %0, %1, off"
                 :: "v"(lds_byte_off), "v"(gaddr) : "memory");
}
__device__ __forceinline__ void wait_asynccnt0() {
    asm volatile("s_wait_asynccnt 0" ::: "memory");
}

// ============================================================================
// Kernel 1: per (b,n) wave -> attention logits (WMMA GEMM), softmax over K=12,
// weighted neighbor aggregation.  agg[BN,128] written to workspace.
// ============================================================================
__global__ void __launch_bounds__(256)
gat_alpha_agg_kernel(const float* __restrict__ neighbor,  // [BN,12,128]
                     const float* __restrict__ nweight,   // [BN,12]
                     const float* __restrict__ extra,     // [BN,128]
                     const float* __restrict__ w1,        // [129,128] row-major
                     const float* __restrict__ w2,        // [128]
                     float* __restrict__ agg_out)         // [BN,128] (workspace)
{
    extern __shared__ float smem[];
    const int tid  = threadIdx.x;
    const int lane = tid & 31;
    const int warp = tid >> 5;

    const unsigned wave_base = W1_FLOATS + warp * WAVE_FLOATS;   // floats
    float* w1_lds = smem;
    float* nb_lds = smem + wave_base;                 // [16][NB_STRIDE]
    float* ex_lds = nb_lds + NB_FLOATS;               // [132]  (128 data, then 1,0,0,0)
    float* w2_lds = ex_lds + 132;                     // [128]

    const int bn = blockIdx.x * K1_WAVES + warp;
    const float* nb_g = neighbor + (size_t)bn * (KNB * DIM);

    // ---- async-stage w1 once per block: 129 rows x 32 b128 chunks ----
    for (int q = tid; q < 129 * 32; q += 256) {
        int r = q >> 5, c4 = (q & 31) * 4;
        async_ld_b128((unsigned)(r * W1_STRIDE + c4) * 4u, w1 + r * 128 + c4);
    }
    for (int idx = tid; idx < 3 * W1_STRIDE; idx += 256)   // zero rows 129..131
        w1_lds[129 * W1_STRIDE + idx] = 0.0f;

    // ---- async-stage this wave's neighbor tile: 12 rows x 32 b128 chunks ----
    for (int q = lane; q < KNB * 32; q += 32) {
        int k = q >> 5, c4 = (q & 31) * 4;
        async_ld_b128((wave_base + (unsigned)(k * NB_STRIDE + c4)) * 4u,
                      nb_g + k * 128 + c4);
    }
    // pad cols 128..131 for rows 0..15 (col128 = edge weight)
    for (int i = lane; i < 16 * 4; i += 32) {
        int k = i >> 2, c = 128 + (i & 3);
        float v = 0.0f;
        if (c == 128 && k < KNB) v = nweight[(size_t)bn * KNB + k];
        nb_lds[k * NB_STRIDE + c] = v;
    }
    // zero rows 12..15 cols 0..127
    for (int i = lane; i < 4 * DIM; i += 32) {
        int k = 12 + (i >> 7), c = i & 127;
        nb_lds[k * NB_STRIDE + c] = 0.0f;
    }
    // extra (one b128 per lane covers all 128 floats) + bias slot {1,0,0,0}
    async_ld_b128((wave_base + NB_FLOATS + (unsigned)(lane * 4)) * 4u,
                  extra + (size_t)bn * DIM + lane * 4);
    if (lane < 4) ex_lds[DIM + lane] = (lane == 0) ? 1.0f : 0.0f;
    // w2
    async_ld_b128((wave_base + NB_FLOATS + 132 + (unsigned)(lane * 4)) * 4u,
                  w2 + lane * 4);

    wait_asynccnt0();
    __syncthreads();

    const int  half = (lane < 16) ? 0 : 2;   // K-offset within 4-wide fragment
    const int  l15  = lane & 15;

    // ---- alpha[16,128] = feat[16,132] @ w1[132,128] via v_wmma_f32_16x16x4 ----
    const v8f vzero = {0.f, 0.f, 0.f, 0.f, 0.f, 0.f, 0.f, 0.f};
    v8f acc[8];
    #pragma unroll
    for (int t = 0; t < 8; ++t) acc[t] = vzero;

    #pragma unroll 1
    for (int kk = 0; kk < 33; ++kk) {        // K = 132 in steps of 4
        const int c0 = 4 * kk + half;
        // batch ALL fragment loads first, so the DS pipe has 10 loads in
        // flight behind one wait, then fire the 8-WMMA burst back-to-back
        float2 nbp = *(const float2*)&nb_lds[l15 * NB_STRIDE + c0];
        float2 exv = *(const float2*)&ex_lds[c0];
        v2f bfr[8];
        #pragma unroll
        for (int t = 0; t < 8; ++t) {
            bfr[t].x = w1_lds[(c0    ) * W1_STRIDE + t * 16 + l15];
            bfr[t].y = w1_lds[(c0 + 1) * W1_STRIDE + t * 16 + l15];
        }
        v2f a = { nbp.x * exv.x, nbp.y * exv.y };   // feat = extra ⊙ neighbor
        #pragma unroll
        for (int t = 0; t < 8; ++t)
            acc[t] = wmma_f32_4(a, bfr[t], acc[t]);
    }

    // ---- leaky-ReLU + w2 projection: per-lane partial over its 8 columns ----
    float S[8];
    #pragma unroll
    for (int j = 0; j < 8; ++j) S[j] = 0.0f;
    #pragma unroll
    for (int t = 0; t < 8; ++t) {
        float w2v = w2_lds[t * 16 + l15];
        #pragma unroll
        for (int j = 0; j < 8; ++j) {
            float a = acc[t][j];
            a = (a > 0.0f) ? a : 0.2f * a;
            S[j] += a * w2v;
        }
    }
    // butterfly within each 16-lane half (rows j for lanes<16, rows j+8 for lanes>=16)
    #pragma unroll
    for (int off = 1; off < 16; off <<= 1) {
        #pragma unroll
        for (int j = 0; j < 8; ++j) S[j] += __shfl_xor(S[j], off, 32);
    }
    float T[8];
    #pragma unroll
    for (int j = 0; j < 8; ++j) T[j] = __shfl_xor(S[j], 16, 32);

    float logit[12];
    #pragma unroll
    for (int j = 0; j < 8; ++j)  logit[j] = (lane < 16) ? S[j] : T[j];
    #pragma unroll
    for (int j = 8; j < 12; ++j) logit[j] = (lane < 16) ? T[j - 8] : S[j - 8];

    // ---- softmax over K=12 (identical in every lane) ----
    float m = logit[0];
    #pragma unroll
    for (int j = 1; j < 12; ++j) m = fmaxf(m, logit[j]);
    float p[12], sum = 0.0f;
    #pragma unroll
    for (int j = 0; j < 12; ++j) { p[j] = __expf(logit[j] - m); sum += p[j]; }
    const float inv = 1.0f / sum;

    // ---- agg[d] = sum_k p[k] * neighbor[k][d]; each lane owns 4 columns ----
    float4 agg = {0.f, 0.f, 0.f, 0.f};
    #pragma unroll
    for (int k = 0; k < 12; ++k) {
        float4 nbv = *(const float4*)&nb_lds[k * NB_STRIDE + 4 * lane];
        float pk = p[k] * inv;
        agg.x += pk * nbv.x; agg.y += pk * nbv.y;
        agg.z += pk * nbv.z; agg.w += pk * nbv.w;
    }
    *(float4*)&agg_out[(size_t)bn * DIM + 4 * lane] = agg;
}

// ============================================================================
// Kernel 2: out[32768,128] = relu( [self||agg][32768,256] @ w3[256,128] )
// One block per 16-row M-tile (A async-staged to LDS), one wave per 16-col N-tile.
// ============================================================================
__global__ void __launch_bounds__(256)
gat_out_kernel(const float* __restrict__ selfv,  // [BN,128]
               const float* __restrict__ agg,    // [BN,128]
               const float* __restrict__ w3,     // [256,128] row-major
               float* __restrict__ out)          // [BN,128]
{
    extern __shared__ float smem[];              // [16][K2_A_STRIDE]
    const int tid  = threadIdx.x;
    const int lane = tid & 31;
    const int warp = tid >> 5;
    const int m0   = blockIdx.x * 16;

    // async-stage A tile: 16 rows, self -> cols 0..127, agg -> cols 128..255
    for (int q = tid; q < 16 * 32; q += 256) {
        int r = q >> 5, c4 = (q & 31) * 4;
        async_ld_b128((unsigned)(r * K2_A_STRIDE + c4) * 4u,
                      selfv + (size_t)(m0 + r) * 128 + c4);
    }
    for (int q = tid; q < 16 * 32; q += 256) {
        int r = q >> 5, c4 = (q & 31) * 4;
        async_ld_b128((unsigned)(r * K2_A_STRIDE + 128 + c4) * 4u,
                      agg + (size_t)(m0 + r) * 128 + c4);
    }
    wait_asynccnt0();
    __syncthreads();

    const int n0   = warp * 16;
    const int half = (lane < 16) ? 0 : 2;
    const int l15  = lane & 15;

    v8f acc = {0.f, 0.f, 0.f, 0.f, 0.f, 0.f, 0.f, 0.f};
    #pragma unroll 1
    for (int ks4 = 0; ks4 < 16; ++ks4) {         // K = 256, 4 WMMA steps per iter
        v2f av[4], bv[4];
        #pragma unroll
        for (int u = 0; u < 4; ++u) {
            const int c0 = 4 * (ks4 * 4 + u) + half;
            float2 ap = *(const float2*)&smem[l15 * K2_A_STRIDE + c0];
            av[u].x = ap.x; av[u].y = ap.y;
            bv[u].x = w3[(size_t)(c0    ) * 128 + n0 + l15];
            bv[u].y = w3[(size_t)(c0 + 1) * 128 + n0 + l15];
        }
        #pragma unroll
        for (int u = 0; u < 4; ++u)
            acc = wmma_f32_4(av[u], bv[u], acc);
    }

    #pragma unroll
    for (int j = 0; j < 8; ++j) {
        float v = acc[j];
        v = (v > 0.0f) ? v : 0.0f;
        out[(size_t)(m0 + j + ((lane < 16) ? 0 : 8)) * 128 + n0 + l15] = v;
    }
}

// ============================================================================
extern "C" void kernel_launch(void* const* d_in, const int* in_sizes, int n_in,
                              void* d_out, int out_size, void* d_ws, size_t ws_size,
                              hipStream_t stream) {
    const float* selfv    = (const float*)d_in[0];  // [B,N,128]
    const float* neighbor = (const float*)d_in[1];  // [B,N,12,128]
    // d_in[2] = batch_size (scalar, unused), d_in[3] = masks (unused by reference)
    const float* nweight  = (const float*)d_in[4];  // [B,N,12]
    const float* extra    = (const float*)d_in[5];  // [B,N,128]
    const float* w1       = (const float*)d_in[6];  // [129,128]
    const float* w2       = (const float*)d_in[7];  // [128,1]
    const float* w3       = (const float*)d_in[8];  // [256,128]
    float*       outp     = (float*)d_out;
    float*       aggws    = (float*)d_ws;           // 32768*128 floats = 16.8 MB scratch

    const size_t lds1 = (size_t)K1_LDS_FLOATS * sizeof(float);   // ~145.6 KB
    gat_alpha_agg_kernel<<<BN / K1_WAVES, 256, lds1, stream>>>(
        neighbor, nweight, extra, w1, w2, aggws);

    const size_t lds2 = (size_t)K2_LDS_FLOATS * sizeof(float);   // ~16.6 KB
    gat_out_kernel<<<BN / 16, 256, lds2, stream>>>(selfv, aggws, w3, outp);
}